// GraphSAGE_18657337933835
// MI455X (gfx1250) — compile-verified
//
#include <hip/hip_runtime.h>
#include <math.h>

// Problem constants (match reference)
#define NNODES 50000
#define NEDGES 800000
#define FDIM   128
#define HDIM   128
#define NGRAPH 64

// Swizzled-weight LDS layout: fragment for (n-tile t, lane, k-chunk c, elem j)
// lives at  t*LTILE + lane*66 + c*2 + j  (floats).
// - per-lane stride 66 floats (264B): 8B-aligned ds_load_b64, banks (2*lane+2c)%64
//   -> 32 distinct banks, conflict-free
// - per-tile stride 2112 floats (8448B): immediate ds offsets up to 59KB < 64KB
#define LTILE 2112
#define LMAT  (8 * LTILE)   // 16896 floats per 128x128 matrix

typedef __attribute__((ext_vector_type(2))) float v2f;
typedef __attribute__((ext_vector_type(8))) float v8f;

// Non-returning fp32 atomic add -> global_atomic_add_f32 (no CAS loop)
__device__ __forceinline__ void atomAddF(float* p, float v) {
    unsafeAtomicAdd(p, v);
}

// ---------------- utility: zero fill ----------------
__global__ void fill_zero(float* __restrict__ p, int n) {
    int i = blockIdx.x * blockDim.x + threadIdx.x;
    int stride = gridDim.x * blockDim.x;
    for (; i < n; i += stride) p[i] = 0.0f;
}

// ---------------- in-degree counts (as float) ----------------
__global__ void count_edges(const int* __restrict__ dst, float* __restrict__ cnt, int E) {
    int e = blockIdx.x * blockDim.x + threadIdx.x;
    if (e < E) atomAddF(&cnt[dst[e]], 1.0f);
}

// ---------------- scatter-add of feature rows: msum[dst] += x[src] ----------------
// One wave per edge: 32 lanes x float4 = 128 floats = one 512B row (coalesced gather).
__global__ void scatter_rows(const float* __restrict__ x, const int* __restrict__ src,
                             const int* __restrict__ dst, float* __restrict__ msum, int E) {
    int t = blockIdx.x * blockDim.x + threadIdx.x;
    int e = t >> 5;
    int l = t & 31;
    if (e >= E) return;
    int s = src[e];
    int d = dst[e];
    const float4* xr = (const float4*)(x + (long long)s * FDIM);
    float4 v = xr[l];
    float* o = msum + (long long)d * FDIM + l * 4;
    atomAddF(o + 0, v.x);
    atomAddF(o + 1, v.y);
    atomAddF(o + 2, v.z);
    atomAddF(o + 3, v.w);
}

// ---------------- global add pool: pool[batch[node]] += h[node] ----------------
__global__ void pool_rows(const float* __restrict__ h, const int* __restrict__ batch,
                          float* __restrict__ pool, int n) {
    int t = blockIdx.x * blockDim.x + threadIdx.x;
    int node = t >> 5;
    int l = t & 31;
    if (node >= n) return;
    int g = batch[node];
    const float4* hr = (const float4*)(h + (long long)node * HDIM);
    float4 v = hr[l];
    float* o = pool + (long long)g * HDIM + l * 4;
    atomAddF(o + 0, v.x);
    atomAddF(o + 1, v.y);
    atomAddF(o + 2, v.z);
    atomAddF(o + 3, v.w);
}

// ---------------- SAGE layer via fp32 WMMA ----------------
// out = relu( (msum/max(cnt,1)) @ Wl^T + xin @ Wr^T + bias ), all [N,128] fp32.
// Block = 256 threads = 8 waves; wave w owns a 32-row node tile (two 16x16 M-tiles,
// 2x8 v8f accumulators) and all 8 column tiles of H=128. Each B fragment (ds_load_b64
// with immediate offset from the swizzled LDS image) feeds 2 WMMAs; each (mean,x)
// A-fragment pair feeds 8. A fragments are software-pipelined one k-chunk ahead
// (global), B fragments one n-tile ahead (LDS).
__global__ void __launch_bounds__(256)
sage_wmma(const float* __restrict__ msum, const float* __restrict__ cnt,
          const float* __restrict__ xin,
          const float* __restrict__ Wl, const float* __restrict__ Wr,
          const float* __restrict__ bias, float* __restrict__ out, int n) {
    extern __shared__ float smem[];
    float* ldsL = smem;
    float* ldsR = smem + LMAT;

    int tid = threadIdx.x;
    // Swizzled fill: consecutive threads read consecutive W elements (coalesced);
    // decompose linear idx -> (t, lane, c, j) fragment coordinates.
    for (int idx = tid; idx < HDIM * FDIM; idx += 256) {
        int col = idx & 127;
        int row = idx >> 7;
        int t   = row >> 4;
        int lr  = row & 15;
        int c   = col >> 2;
        int kj  = col & 3;
        int ln  = lr + ((kj >> 1) << 4);
        int off = t * LTILE + ln * 66 + c * 2 + (kj & 1);
        ldsL[off] = Wl[idx];
        ldsR[off] = Wr[idx];
    }
    __syncthreads();

    int wave = tid >> 5;
    int lane = tid & 31;
    int m0   = blockIdx.x * 256 + wave * 32;   // 32 rows per wave
    int lm   = lane & 15;
    int k0   = lane >> 4;                      // v2f index offset: koff/2

    // Clamp load rows so EXEC stays all-ones across WMMA (stores are guarded instead).
    int r0 = m0 + lm;
    int r1 = m0 + 16 + lm;
    int c0 = (r0 < n) ? r0 : (n - 1);
    int c1 = (r1 < n) ? r1 : (n - 1);

    float inv0 = 1.0f / fmaxf(cnt[c0], 1.0f);
    float inv1 = 1.0f / fmaxf(cnt[c1], 1.0f);

    const v2f* m0p = (const v2f*)(msum + (long long)c0 * FDIM);
    const v2f* m1p = (const v2f*)(msum + (long long)c1 * FDIM);
    const v2f* x0p = (const v2f*)(xin  + (long long)c0 * FDIM);
    const v2f* x1p = (const v2f*)(xin  + (long long)c1 * FDIM);

    const float* bL = ldsL + lane * 66;   // per-lane fragment base (matrix L)
    const float* bR = ldsR + lane * 66;   // per-lane fragment base (matrix R)

    v8f accA[8] = {};   // rows m0 .. m0+15
    v8f accB[8] = {};   // rows m0+16 .. m0+31

    // Software-pipelined A fragments (one k-chunk ahead)
    v2f am0n = m0p[k0];
    v2f am1n = m1p[k0];
    v2f ax0n = x0p[k0];
    v2f ax1n = x1p[k0];

    for (int c = 0; c < 32; ++c) {             // k-chunk (K = 4*c)
        v2f am0 = am0n, am1 = am1n, ax0 = ax0n, ax1 = ax1n;
        int cn = ((c + 1) & 31) * 2 + k0;      // wraps to chunk 0 on last iter
        am0n = m0p[cn];
        am1n = m1p[cn];
        ax0n = x0p[cn];
        ax1n = x1p[cn];

        am0.x *= inv0; am0.y *= inv0;
        am1.x *= inv1; am1.y *= inv1;

        // B fragments pipelined one n-tile ahead; all ds_load_b64 immediate-offset
        v2f bl = *(const v2f*)(bL + c * 2);
        v2f br = *(const v2f*)(bR + c * 2);
#pragma unroll
        for (int t = 0; t < 8; ++t) {
            int tn = (t + 1) & 7;
            v2f bln = *(const v2f*)(bL + tn * LTILE + c * 2);
            v2f brn = *(const v2f*)(bR + tn * LTILE + c * 2);
            accA[t] = __builtin_amdgcn_wmma_f32_16x16x4_f32(
                false, am0, false, bl, (short)0, accA[t], false, false);
            accB[t] = __builtin_amdgcn_wmma_f32_16x16x4_f32(
                false, am1, false, bl, (short)0, accB[t], false, false);
            accA[t] = __builtin_amdgcn_wmma_f32_16x16x4_f32(
                false, ax0, false, br, (short)0, accA[t], false, false);
            accB[t] = __builtin_amdgcn_wmma_f32_16x16x4_f32(
                false, ax1, false, br, (short)0, accB[t], false, false);
            bl = bln;
            br = brn;
        }
    }

    // C/D layout: VGPR j -> M = j + 8*(lane>=16), N = lane&15 within each 16x16 tile.
    int rbase = m0 + ((lane >> 4) << 3);
#pragma unroll
    for (int t = 0; t < 8; ++t) {
        int col = t * 16 + lm;
        float bv = bias[col];
#pragma unroll
        for (int j = 0; j < 8; ++j) {
            int rowA = rbase + j;
            if (rowA < n) {
                float v = accA[t][j] + bv;
                out[(long long)rowA * HDIM + col] = (v > 0.0f) ? v : 0.0f;
            }
            int rowB = rbase + 16 + j;
            if (rowB < n) {
                float v = accB[t][j] + bv;
                out[(long long)rowB * HDIM + col] = (v > 0.0f) ? v : 0.0f;
            }
        }
    }
}

// ---------------- fused MLP head + log_softmax (tiny: 64 x 256 input) ----------------
__global__ void mlp_head(const float* __restrict__ scp, const float* __restrict__ fcp,
                         const float* __restrict__ W1, const float* __restrict__ b1,
                         const float* __restrict__ W2, const float* __restrict__ b2,
                         const float* __restrict__ W3, const float* __restrict__ b3,
                         float* __restrict__ out) {
    __shared__ float cat[2 * HDIM];
    __shared__ float h1[HDIM];
    __shared__ float h2[HDIM / 2];
    __shared__ float lg[2];
    int g = blockIdx.x;
    int tid = threadIdx.x;   // 128 threads

    cat[tid]        = scp[g * HDIM + tid];
    cat[tid + HDIM] = fcp[g * HDIM + tid];
    __syncthreads();

    float s = b1[tid];
    const float* w = W1 + tid * (2 * HDIM);
    for (int k = 0; k < 2 * HDIM; ++k) s += cat[k] * w[k];
    h1[tid] = fmaxf(s, 0.0f);
    __syncthreads();

    if (tid < HDIM / 2) {
        float s2 = b2[tid];
        const float* w2 = W2 + tid * HDIM;
        for (int k = 0; k < HDIM; ++k) s2 += h1[k] * w2[k];
        h2[tid] = fmaxf(s2, 0.0f);
    }
    __syncthreads();

    if (tid < 2) {
        float s3 = b3[tid];
        const float* w3 = W3 + tid * (HDIM / 2);
        for (int k = 0; k < HDIM / 2; ++k) s3 += h2[k] * w3[k];
        lg[tid] = s3;
    }
    __syncthreads();

    if (tid == 0) {
        float m = fmaxf(lg[0], lg[1]);
        float z = expf(lg[0] - m) + expf(lg[1] - m);
        float lse = m + logf(z);
        out[g * 2 + 0] = lg[0] - lse;
        out[g * 2 + 1] = lg[1] - lse;
    }
}

extern "C" void kernel_launch(void* const* d_in, const int* in_sizes, int n_in,
                              void* d_out, int out_size, void* d_ws, size_t ws_size,
                              hipStream_t stream) {
    (void)in_sizes; (void)n_in; (void)out_size; (void)ws_size;

    const float* sc_x  = (const float*)d_in[0];
    const float* fc_x  = (const float*)d_in[1];
    const int*   sc_ei = (const int*)d_in[2];
    const int*   fc_ei = (const int*)d_in[3];
    const int*   batch = (const int*)d_in[4];
    const float* sc_Wl1 = (const float*)d_in[5];
    const float* sc_Wr1 = (const float*)d_in[6];
    const float* sc_b1  = (const float*)d_in[7];
    const float* sc_Wl2 = (const float*)d_in[8];
    const float* sc_Wr2 = (const float*)d_in[9];
    const float* sc_b2  = (const float*)d_in[10];
    const float* fc_Wl1 = (const float*)d_in[11];
    const float* fc_Wr1 = (const float*)d_in[12];
    const float* fc_b1  = (const float*)d_in[13];
    const float* fc_Wl2 = (const float*)d_in[14];
    const float* fc_Wr2 = (const float*)d_in[15];
    const float* fc_b2  = (const float*)d_in[16];
    const float* W1  = (const float*)d_in[17];
    const float* bh1 = (const float*)d_in[18];
    const float* W2  = (const float*)d_in[19];
    const float* bh2 = (const float*)d_in[20];
    const float* W3  = (const float*)d_in[21];
    const float* bh3 = (const float*)d_in[22];

    // Workspace layout (floats): msum | hA | hB | cnt | sc_pool | fc_pool  (~77 MB)
    float* ws = (float*)d_ws;
    const size_t NH = (size_t)NNODES * HDIM;
    float* msum = ws;
    float* hA   = msum + NH;
    float* hB   = hA + NH;
    float* cnt  = hB + NH;
    float* scp  = cnt + NNODES;
    float* fcp  = scp + (size_t)NGRAPH * HDIM;

    const size_t ldsBytes    = (size_t)2 * LMAT * sizeof(float); // ~135 KB of 320 KB/WGP
    const int gemmBlocks     = (NNODES + 255) / 256;             // 32 rows per wave, 8 waves
    const int scatterBlocks  = (NEDGES * 32 + 255) / 256;
    const int poolBlocks     = (NNODES * 32 + 255) / 256;
    const int countBlocks    = (NEDGES + 255) / 256;

    struct Br {
        const float* x; const int* ei;
        const float *Wl1, *Wr1, *b1, *Wl2, *Wr2, *b2;
        float* pool;
    };
    Br br[2] = {
        { sc_x, sc_ei, sc_Wl1, sc_Wr1, sc_b1, sc_Wl2, sc_Wr2, sc_b2, scp },
        { fc_x, fc_ei, fc_Wl1, fc_Wr1, fc_b1, fc_Wl2, fc_Wr2, fc_b2, fcp },
    };

    for (int i = 0; i < 2; ++i) {
        const int* src = br[i].ei;
        const int* dst = br[i].ei + NEDGES;

        fill_zero<<<256, 256, 0, stream>>>(cnt, NNODES);
        count_edges<<<countBlocks, 256, 0, stream>>>(dst, cnt, NEDGES);

        // Layer 1
        fill_zero<<<2048, 256, 0, stream>>>(msum, (int)NH);
        scatter_rows<<<scatterBlocks, 256, 0, stream>>>(br[i].x, src, dst, msum, NEDGES);
        sage_wmma<<<gemmBlocks, 256, ldsBytes, stream>>>(
            msum, cnt, br[i].x, br[i].Wl1, br[i].Wr1, br[i].b1, hA, NNODES);

        // Layer 2
        fill_zero<<<2048, 256, 0, stream>>>(msum, (int)NH);
        scatter_rows<<<scatterBlocks, 256, 0, stream>>>(hA, src, dst, msum, NEDGES);
        sage_wmma<<<gemmBlocks, 256, ldsBytes, stream>>>(
            msum, cnt, hA, br[i].Wl2, br[i].Wr2, br[i].b2, hB, NNODES);

        // Pool
        fill_zero<<<64, 256, 0, stream>>>(br[i].pool, NGRAPH * HDIM);
        pool_rows<<<poolBlocks, 256, 0, stream>>>(hB, batch, br[i].pool, NNODES);
    }

    mlp_head<<<NGRAPH, HDIM, 0, stream>>>(scp, fcp, W1, bh1, W2, bh2, W3, bh3, (float*)d_out);
}